// GCN_43585328120189
// MI455X (gfx1250) — compile-verified
//
#include <hip/hip_runtime.h>
#include <hip/hip_bf16.h>

typedef __attribute__((ext_vector_type(2))) float v2f;
typedef __attribute__((ext_vector_type(8))) float v8f;

#define FEAT 128
#define NCLS 10
#define NGRAPH 64

// ---------------------------------------------------------------------------
// Fused GEMM + self-loop epilogue:
//   H[M x 128]   = A[M x 128] @ W[128 x 128]          (fp32 WMMA 16x16x4)
//   O[M x 128]   = H * dis[row]^2                     (scatter-add init)
// 256 threads = 8 waves; W staged pair-interleaved in LDS (64 KB) so each
// B fragment is a single aligned ds_load_b64. Wave w handles 16 rows.
// ---------------------------------------------------------------------------
__global__ __launch_bounds__(256) void gcn_gemm128_fused(
    const float* __restrict__ A, const float* __restrict__ W,
    const float* __restrict__ dis, float* __restrict__ H,
    float* __restrict__ O, int M) {
  // pair-interleaved: element (k,c) at ((k>>1)*128 + c)*2 + (k&1)
  __shared__ float Wl[64 * FEAT * 2];  // 64 KB

  const int tid = threadIdx.x;
  const float4* __restrict__ Wv = (const float4*)W;
#pragma unroll
  for (int i = 0; i < 16; ++i) {
    const int f = tid + i * 256;   // float4 index into W
    const int k = f >> 5;          // K row 0..127
    const int c = (f & 31) * 4;    // col base
    const float4 v = Wv[f];
    const int base = ((k >> 1) * FEAT + c) * 2 + (k & 1);
    Wl[base + 0] = v.x;
    Wl[base + 2] = v.y;
    Wl[base + 4] = v.z;
    Wl[base + 6] = v.w;
  }
  __syncthreads();

  const int wave     = tid >> 5;
  const int lane     = tid & 31;
  const int laneHalf = lane >> 4;  // 0: lanes 0-15, 1: lanes 16-31
  const int lane16   = lane & 15;
  const int rowBase  = blockIdx.x * 128 + wave * 16;

  v8f acc[8] = {};  // 8 N-tiles of 16x16 f32

  // A fragment row: clamp instead of predicating (tail discarded at store)
  const int arow = min(rowBase + lane16, M - 1);
  const float* __restrict__ Arow = A + (size_t)arow * FEAT;

#pragma unroll 2
  for (int k0 = 0; k0 < FEAT; k0 += 4) {
    const int kk = k0 + laneHalf * 2;            // even
    const v2f a = *(const v2f*)(Arow + kk);      // one global_load_b64
    const float* __restrict__ Wp = Wl + (kk >> 1) * (FEAT * 2);
#pragma unroll
    for (int t = 0; t < 8; ++t) {
      const int col = t * 16 + lane16;
      const v2f b = *(const v2f*)(Wp + col * 2);  // one ds_load_b64
      acc[t] = __builtin_amdgcn_wmma_f32_16x16x4_f32(
          false, a, false, b, (short)0, acc[t], false, false);
    }
  }

  // Epilogue: VGPR r holds (M = rowBase + r + laneHalf*8, N = t*16 + lane16)
#pragma unroll
  for (int r = 0; r < 8; ++r) {
    const int row = rowBase + r + laneHalf * 8;
    const bool ok = row < M;
    const float dv = dis[ok ? row : 0];
    const float w = dv * dv;
    const size_t rb = (size_t)row * FEAT;
#pragma unroll
    for (int t = 0; t < 8; ++t) {
      const int col = t * 16 + lane16;
      if (ok) {
        const float h = acc[t][r];
        H[rb + col] = h;
        O[rb + col] = h * w;  // self-loop message init
      }
    }
  }
}

// ---------------------------------------------------------------------------
// Degree / normalization
// ---------------------------------------------------------------------------
__global__ void deg_init(float* __restrict__ deg, int n) {
  int i = blockIdx.x * blockDim.x + threadIdx.x;
  if (i < n) deg[i] = 1.0f;  // self-loop
}
__global__ void deg_edges(const int* __restrict__ dst, float* __restrict__ deg,
                          int e) {
  int i = blockIdx.x * blockDim.x + threadIdx.x;
  if (i < e) atomicAdd(&deg[dst[i]], 1.0f);
}
__global__ void deg_to_dis(float* __restrict__ deg, int n) {
  int i = blockIdx.x * blockDim.x + threadIdx.x;
  if (i < n) deg[i] = rsqrtf(deg[i]);  // deg >= 1 always (self-loops)
}

// ---------------------------------------------------------------------------
// Edge scatter: one wave per edge, lane handles a float4 of the feature row
// ---------------------------------------------------------------------------
__global__ void edge_scatter(const float* __restrict__ h,
                             const int* __restrict__ src,
                             const int* __restrict__ dst,
                             const float* __restrict__ dis,
                             float* __restrict__ out, int e) {
  int wid = (blockIdx.x * blockDim.x + threadIdx.x) >> 5;
  int lane = threadIdx.x & 31;
  if (wid >= e) return;
  int s = src[wid], d = dst[wid];
  float w = dis[s] * dis[d];
  float4 v = ((const float4*)(h + (size_t)s * FEAT))[lane];
  float* o = out + (size_t)d * FEAT + lane * 4;
  atomicAdd(o + 0, v.x * w);
  atomicAdd(o + 1, v.y * w);
  atomicAdd(o + 2, v.z * w);
  atomicAdd(o + 3, v.w * w);
}

// ---------------------------------------------------------------------------
// out = relu(out + b), in place, one thread per float4
// ---------------------------------------------------------------------------
__global__ void bias_relu(float* __restrict__ out, const float* __restrict__ b,
                          int n) {
  int t = blockIdx.x * blockDim.x + threadIdx.x;
  int node = t >> 5, j = t & 31;
  if (node >= n) return;
  float4 v = ((float4*)(out + (size_t)node * FEAT))[j];
  float4 bb = ((const float4*)b)[j];
  v.x = fmaxf(v.x + bb.x, 0.0f);
  v.y = fmaxf(v.y + bb.y, 0.0f);
  v.z = fmaxf(v.z + bb.z, 0.0f);
  v.w = fmaxf(v.w + bb.w, 0.0f);
  ((float4*)(out + (size_t)node * FEAT))[j] = v;
}

// ---------------------------------------------------------------------------
// Pooling
// ---------------------------------------------------------------------------
__global__ void pool_zero(float* __restrict__ sums, float* __restrict__ cnts) {
  int i = blockIdx.x * blockDim.x + threadIdx.x;
  if (i < NGRAPH * FEAT) sums[i] = 0.0f;
  if (i < NGRAPH) cnts[i] = 0.0f;
}
__global__ void pool_acc(const float* __restrict__ h,
                         const int* __restrict__ batch,
                         float* __restrict__ sums, float* __restrict__ cnts,
                         int n) {
  int t = blockIdx.x * blockDim.x + threadIdx.x;
  int node = t >> 5, lane = t & 31;
  if (node >= n) return;
  int g = batch[node];
  float4 v = ((const float4*)(h + (size_t)node * FEAT))[lane];
  float* s = sums + (size_t)g * FEAT + lane * 4;
  atomicAdd(s + 0, v.x);
  atomicAdd(s + 1, v.y);
  atomicAdd(s + 2, v.z);
  atomicAdd(s + 3, v.w);
  if (lane == 0) atomicAdd(&cnts[g], 1.0f);
}

// Final FC: out[g,c] = (sum_k sums[g,k]*Wfc[k,c]) / max(cnt,1) + bfc[c]
__global__ void fc_out(const float* __restrict__ sums,
                       const float* __restrict__ cnts,
                       const float* __restrict__ Wfc,
                       const float* __restrict__ bfc, float* __restrict__ out) {
  int t = blockIdx.x * blockDim.x + threadIdx.x;
  if (t >= NGRAPH * NCLS) return;
  int g = t / NCLS, c = t % NCLS;
  float inv = 1.0f / fmaxf(cnts[g], 1.0f);
  float acc = 0.0f;
#pragma unroll 8
  for (int k = 0; k < FEAT; ++k) acc += sums[g * FEAT + k] * Wfc[k * NCLS + c];
  out[t] = acc * inv + bfc[c];
}

// ---------------------------------------------------------------------------
extern "C" void kernel_launch(void* const* d_in, const int* in_sizes, int n_in,
                              void* d_out, int out_size, void* d_ws,
                              size_t ws_size, hipStream_t stream) {
  const float* x    = (const float*)d_in[0];
  const int*   eidx = (const int*)d_in[1];  // [2, E] flat: row0=src, row1=dst
  const int*   bat  = (const int*)d_in[2];
  const float* W1   = (const float*)d_in[3];
  const float* b1   = (const float*)d_in[4];
  const float* W2   = (const float*)d_in[5];
  const float* b2   = (const float*)d_in[6];
  const float* Wfc  = (const float*)d_in[7];
  const float* bfc  = (const float*)d_in[8];
  float* out = (float*)d_out;

  const int n = in_sizes[0] / FEAT;   // 100000 nodes
  const int e = in_sizes[1] / 2;      // 1600000 edges
  const int* src = eidx;
  const int* dst = eidx + e;

  // workspace layout (256B aligned)
  char* ws = (char*)d_ws;
  const size_t featBytes = (size_t)n * FEAT * sizeof(float);  // 51.2 MB
  float* bufA = (float*)(ws);                                  // h
  float* bufB = (float*)(ws + featBytes);                      // aggregated
  float* dis  = (float*)(ws + 2 * featBytes);                  // deg -> dis
  size_t disPad = (((size_t)n * sizeof(float)) + 255) & ~(size_t)255;
  float* sums = (float*)(ws + 2 * featBytes + disPad);
  float* cnts = (float*)((char*)sums + NGRAPH * FEAT * sizeof(float));

  const int nodeBlk    = (n + 255) / 256;
  const int edgeBlk    = (e + 255) / 256;
  const int featVecBlk = (n * 32 + 255) / 256;   // one thread per float4 chunk
  const int waveEdgeBlk = (int)(((size_t)e * 32 + 255) / 256);
  const int gemmBlk    = (n + 127) / 128;

  // --- normalization: deg (with self loop) -> dis = rsqrt(deg) ---
  deg_init<<<nodeBlk, 256, 0, stream>>>(dis, n);
  deg_edges<<<edgeBlk, 256, 0, stream>>>(dst, dis, e);
  deg_to_dis<<<nodeBlk, 256, 0, stream>>>(dis, n);

  // --- layer 1: h1 = x@W1 (+self-loop init), scatter, bias+relu ---
  gcn_gemm128_fused<<<gemmBlk, 256, 0, stream>>>(x, W1, dis, bufA, bufB, n);
  edge_scatter<<<waveEdgeBlk, 256, 0, stream>>>(bufA, src, dst, dis, bufB, e);
  bias_relu<<<featVecBlk, 256, 0, stream>>>(bufB, b1, n);

  // --- layer 2 ---
  gcn_gemm128_fused<<<gemmBlk, 256, 0, stream>>>(bufB, W2, dis, bufA, bufB, n);
  edge_scatter<<<waveEdgeBlk, 256, 0, stream>>>(bufA, src, dst, dis, bufB, e);
  bias_relu<<<featVecBlk, 256, 0, stream>>>(bufB, b2, n);

  // --- mean pool + FC ---
  pool_zero<<<(NGRAPH * FEAT + 255) / 256, 256, 0, stream>>>(sums, cnts);
  pool_acc<<<featVecBlk, 256, 0, stream>>>(bufB, bat, sums, cnts, n);
  fc_out<<<(NGRAPH * NCLS + 255) / 256, 256, 0, stream>>>(sums, cnts, Wfc, bfc,
                                                          out);
}